// Blur_29068338659488
// MI455X (gfx1250) — compile-verified
//
#include <hip/hip_runtime.h>

// ---------------------------------------------------------------------------
// upfirdn2d up=2, pad=(2,1,2,1), separable filter outer([1,3,3,1])/16.
// in : (16, 512, 32, 32) f32   ->   out: (16, 512, 64, 64) f32
//
// Roofline: 33.5 MB read + 134 MB write = ~168 MB @ 23.3 TB/s => ~7.2 us.
// Effective compute: 4 MAC/output (zero-insertion kills 12 of 16 taps)
// => 0.27 GFLOP, ~1.6 FLOP/byte. Pure bandwidth kernel: WMMA would only
// add wasted FLOPs + layout shuffles. CDNA5 feature used instead: async
// global->LDS staging (ASYNCcnt) + non-temporal b128 streaming stores.
// ---------------------------------------------------------------------------

typedef float v4f __attribute__((ext_vector_type(4)));
typedef int   v4i __attribute__((ext_vector_type(4)));
typedef __attribute__((address_space(1))) v4i* glob_v4i_ptr;
typedef __attribute__((address_space(3))) v4i* lds_v4i_ptr;

constexpr int H = 32, W = 32, OH = 64, OW = 64;
// LDS tile padded with zeros: rows -1..32 -> 34 rows, cols -4..35 -> 40 cols.
// Col pad of 4 keeps interior 16B-aligned for b128 LDS stores.
constexpr int PR = 34, PC = 40;               // 1360 floats = 340 x 16B

#if defined(__has_builtin)
#if __has_builtin(__builtin_amdgcn_global_load_async_to_lds_b128) && \
    __has_builtin(__builtin_amdgcn_s_wait_asynccnt)
#define USE_ASYNC_LDS 1
#endif
#endif

__global__ __launch_bounds__(256) void upfirdn2d_up2_k1331(
    const float* __restrict__ x, float* __restrict__ out)
{
    __shared__ __align__(16) float tile[PR * PC];
    const int img = blockIdx.x;
    const int tid = threadIdx.x;

    // Branch-free zero fill: 340 b128 chunks over 256 threads (256 + 84).
    const v4f z = {0.f, 0.f, 0.f, 0.f};
    *reinterpret_cast<v4f*>(&tile[tid * 4]) = z;
    if (tid < (PR * PC - 1024) / 4)           // 84 threads
        *reinterpret_cast<v4f*>(&tile[1024 + tid * 4]) = z;
    __syncthreads();

    // Stage the 32x32 image: one 16B chunk per thread, coalesced.
    const float* src = x + (size_t)img * (H * W);
    {
        const int r  = tid >> 3;        // 0..31
        const int c4 = (tid & 7) * 4;   // 0,4,...,28
        float* lds_dst = &tile[(r + 1) * PC + 4 + c4];
#if USE_ASYNC_LDS
        __builtin_amdgcn_global_load_async_to_lds_b128(
            (glob_v4i_ptr)(src + r * W + c4),   // v4i addrspace(1)*
            (lds_v4i_ptr)lds_dst,               // v4i addrspace(3)*
            /*offset=*/0, /*cpol=*/0);
        // Issue next-image prefetch while the async copy is in flight.
        __builtin_prefetch(src + H * W + (tid << 2), 0, 1);
        __builtin_amdgcn_s_wait_asynccnt(0);
#else
        v4f v = __builtin_nontemporal_load(
            reinterpret_cast<const v4f*>(src + r * W + c4));
        *reinterpret_cast<v4f*>(lds_dst) = v;
        __builtin_prefetch(src + H * W + (tid << 2), 0, 1);
#endif
    }
    __syncthreads();

    // Each thread: quad-row qy (0..31), 4 consecutive quad-cols => 2x8 outputs.
    const int qy = tid >> 3;
    const int tx = tid & 7;
    const int J0 = tx * 4;

    // padded col index of input col (J0-1) is 4 + (J0-1) = 3 + J0
    const float* rm = &tile[(qy + 0) * PC + 3 + J0];  // input row qy-1
    const float* r0 = &tile[(qy + 1) * PC + 3 + J0];  // input row qy
    const float* rp = &tile[(qy + 2) * PC + 3 + J0];  // input row qy+1

    float oe[8], oo[8];  // output rows 2*qy and 2*qy+1
#pragma unroll
    for (int j = 0; j < 4; ++j) {
        // Horizontal: even col 2J -> 1*x[J-1] + 3*x[J]; odd -> 3*x[J] + 1*x[J+1]
        float hm_e = fmaf(3.f, rm[j + 1], rm[j]);
        float hm_o = fmaf(3.f, rm[j + 1], rm[j + 2]);
        float h0_e = fmaf(3.f, r0[j + 1], r0[j]);
        float h0_o = fmaf(3.f, r0[j + 1], r0[j + 2]);
        float hp_e = fmaf(3.f, rp[j + 1], rp[j]);
        float hp_o = fmaf(3.f, rp[j + 1], rp[j + 2]);
        // Vertical: even row -> 1*h[I-1] + 3*h[I]; odd -> 3*h[I] + 1*h[I+1]
        oe[2 * j]     = fmaf(3.f, h0_e, hm_e) * 0.0625f;
        oe[2 * j + 1] = fmaf(3.f, h0_o, hm_o) * 0.0625f;
        oo[2 * j]     = fmaf(3.f, h0_e, hp_e) * 0.0625f;
        oo[2 * j + 1] = fmaf(3.f, h0_o, hp_o) * 0.0625f;
    }

    // Streaming (non-temporal) 16B stores; output is write-once.
    float* dst = out + (size_t)img * (OH * OW);
    v4f e0 = {oe[0], oe[1], oe[2], oe[3]};
    v4f e1 = {oe[4], oe[5], oe[6], oe[7]};
    v4f o0 = {oo[0], oo[1], oo[2], oo[3]};
    v4f o1 = {oo[4], oo[5], oo[6], oo[7]};
    __builtin_nontemporal_store(e0, (v4f*)(dst + (2 * qy) * OW + 2 * J0));
    __builtin_nontemporal_store(e1, (v4f*)(dst + (2 * qy) * OW + 2 * J0 + 4));
    __builtin_nontemporal_store(o0, (v4f*)(dst + (2 * qy + 1) * OW + 2 * J0));
    __builtin_nontemporal_store(o1, (v4f*)(dst + (2 * qy + 1) * OW + 2 * J0 + 4));
}

extern "C" void kernel_launch(void* const* d_in, const int* in_sizes, int n_in,
                              void* d_out, int out_size, void* d_ws, size_t ws_size,
                              hipStream_t stream) {
    const float* x = (const float*)d_in[0];
    float* out = (float*)d_out;
    const int n_images = in_sizes[0] / (H * W);  // 16 * 512 = 8192
    upfirdn2d_up2_k1331<<<dim3(n_images), dim3(256), 0, stream>>>(x, out);
}